// MaxwellModel_1614907703530
// MI455X (gfx1250) — compile-verified
//
#include <hip/hip_runtime.h>
#include <stdint.h>

// Maxwell viscoelastic scan: B=16384 independent rows, sequential over T=2048.
// Memory-bound (384 MB @ 23.3 TB/s ~ 16us). Strategy: CDNA5 Tensor Data Mover
// (tensor_load_to_lds) stages 32x32 tiles per wave into LDS with hardware row
// padding (36-float stride), TENSORcnt-tracked triple buffering; transposed
// scan out of LDS; coalesced B128 writeback. Each wave is fully self-contained
// (owns 32 rows + its own LDS slice) -> zero barriers.

#define T_LEN       2048
#define TT          32          // time-tile (floats per row per tile) = 128B
#define ROWS_PER_WG 128         // 4 waves x 32 rows
#define NWAVE       4
#define NBUF        3           // triple buffering
#define LDS_STRIDE  36          // 32 data + 4 pad floats (TDM pad_interval=32dw, pad_amount=4dw)

// float-index offsets inside smem
#define REGION_SZ   (NBUF * NWAVE * 32 * LDS_STRIDE)   // one array's staging region
#define D_BASE      (REGION_SZ)                        // dts region
#define O_BASE      (2 * REGION_SZ)                    // output staging region
#define SMEM_FLOATS (2 * REGION_SZ + NWAVE * 32 * LDS_STRIDE)  // 32256 floats = 129 KB

typedef unsigned int u32x4 __attribute__((ext_vector_type(4)));
typedef int          i32x4 __attribute__((ext_vector_type(4)));
typedef int          i32x8 __attribute__((ext_vector_type(8)));

__global__ __launch_bounds__(ROWS_PER_WG)
void maxwell_scan_kernel(const float* __restrict__ strains,
                         const float* __restrict__ dts,
                         float* __restrict__ out)
{
    __shared__ __align__(16) float smem[SMEM_FLOATS];

    const int tid = threadIdx.x;
    const int l   = tid & 31;                                    // lane id
    const int w   = __builtin_amdgcn_readfirstlane(tid) >> 5;    // wave id (SGPR-uniform)
    const int row0 = blockIdx.x * ROWS_PER_WG + w * 32;          // this wave's first row
    const int ntiles = T_LEN / TT;                               // 64

    // writeback lane mapping: 4 rows x 128B per b128 store, x8 = 32 rows x 128B
    const int ld_r = l >> 3;           // 0..3
    const int ld_c = (l & 7) * 4;      // 0,4,...,28 (float col), 16B aligned

    // low 32 bits of a generic LDS pointer == byte offset into wave's LDS window
    const uint32_t smem_base = (uint32_t)(uintptr_t)(&smem[0]);

    // wave-uniform global base addresses of this wave's 32-row slab
    const uint64_t gbase_s = (uint64_t)(uintptr_t)(strains + (size_t)row0 * T_LEN);
    const uint64_t gbase_d = (uint64_t)(uintptr_t)(dts     + (size_t)row0 * T_LEN);

    // ---- TDM descriptor group 1 (constant across tiles) ----
    // w0: data_size=2 (4B) <<16 | pad_enable <<20 | pad_interval=4 (32 dw) <<22
    //     | pad_amount=3 (4 dw) <<25 ; workgroup_mask[15:0]=0 (not in cluster)
    // tensor_dim0/1 = 1<<22 (huge: OOB clamp never fires; all addresses in-bounds)
    // tile_dim0=32 elems (128B), tile_dim1=32 rows, tile_dim2=0 (2D)
    // tensor_dim0_stride = T_LEN elements
    const uint32_t TD = 1u << 22;
    i32x8 g1;
    g1[0] = (int)((2u << 16) | (1u << 20) | (4u << 22) | (3u << 25));
    g1[1] = (int)((TD & 0xFFFFu) << 16);                            // dim0[15:0]
    g1[2] = (int)(((TD >> 16) & 0xFFFFu) | ((TD & 0xFFFFu) << 16)); // dim0[31:16] | dim1[15:0]
    g1[3] = (int)(((TD >> 16) & 0xFFFFu) | (32u << 16));            // dim1[31:16] | tile_dim0
    g1[4] = 32;                                                     // tile_dim1 | tile_dim2=0
    g1[5] = T_LEN;                                                  // stride0[31:0]
    g1[6] = 0;                                                      // stride0[47:32] | stride1 lo
    g1[7] = 0;
    const i32x4 z4 = {0, 0, 0, 0};                                  // groups 2/3: 2D tensor
    const i32x8 z8 = {0, 0, 0, 0, 0, 0, 0, 0};                      // extra group (clang-23 form)

    // issue the 2 TDM tile loads (strains + dts) for tile tk into buffer b
    auto issue_tile = [&](int tk, int b) {
        const uint32_t lds_s = smem_base + (uint32_t)(((b * NWAVE + w) * 32) * LDS_STRIDE * 4);
        const uint32_t lds_d = lds_s + (uint32_t)(D_BASE * 4);
        const uint64_t ga_s  = gbase_s + (uint64_t)(tk * TT * 4);
        const uint64_t ga_d  = gbase_d + (uint64_t)(tk * TT * 4);
        u32x4 g0;
        g0[0] = 1u;                                                // count=1, user mode
        g0[1] = lds_s;                                             // lds_addr (bytes)
        g0[2] = (uint32_t)ga_s;                                    // global_addr[31:0]
        g0[3] = ((uint32_t)(ga_s >> 32) & 0x01FFFFFFu) | 0x80000000u; // ga[56:32] | type=2
        __builtin_amdgcn_tensor_load_to_lds(g0, g1, z4, z4, z8, 0);
        g0[1] = lds_d;
        g0[2] = (uint32_t)ga_d;
        g0[3] = ((uint32_t)(ga_d >> 32) & 0x01FFFFFFu) | 0x80000000u;
        __builtin_amdgcn_tensor_load_to_lds(g0, g1, z4, z4, z8, 0);
    };

    // prologue: fill the pipeline (3 tiles x 2 TDM ops = 6 outstanding)
    issue_tile(0, 0);
    issue_tile(1, 1);
    issue_tile(2, 2);

    float gamma = 0.0f, prev_eps = 0.0f, prev_dt = 0.0f;

    for (int k = 0; k < ntiles; ++k) {
        // TDM ops complete in order within a wave: wait until tile k's 2 loads
        // are done, leaving later tiles' loads in flight.
        const int rem = ntiles - 1 - k;
        if (rem >= 2)      { asm volatile("s_wait_tensorcnt 4" ::: "memory"); }
        else if (rem == 1) { asm volatile("s_wait_tensorcnt 2" ::: "memory"); }
        else               { asm volatile("s_wait_tensorcnt 0" ::: "memory"); }

        // ---- scan this lane's row through the tile (transposed read from LDS) ----
        const int sbase = (((k % NBUF) * NWAVE + w) * 32 + l) * LDS_STRIDE;
        const float* srow = &smem[sbase];
        const float* drow = &smem[D_BASE + sbase];
        float*       orow = &smem[O_BASE + (w * 32 + l) * LDS_STRIDE];

#pragma unroll
        for (int t = 0; t < TT; ++t) {
            const float eps = srow[t];
            const float dt  = drow[t];
            float o;
            if (k == 0 && t == 0) {
                o = 0.0f;                                  // stresses[:,0] = 0
            } else {
                // gamma_n = gamma + dt[n-1]*(E/eta)*(eps[n-1]-gamma);  E/eta = 2
                gamma = gamma + (prev_dt * 2.0f) * (prev_eps - gamma);
                // sigma = E_inf*eps + E*(eps-gamma) = eps + 2*(eps-gamma)
                o = eps + 2.0f * (eps - gamma);
            }
            prev_eps = eps;
            prev_dt  = dt;
            orow[t]  = o;
        }

        // buffer (k+3)%3 == k%3 is now free (its ds_loads fully consumed): refill
        if (k + 3 < ntiles) issue_tile(k + 3, (k + 3) % NBUF);

        // ---- coalesced writeback of this wave's 32x32 output tile ----
#pragma unroll
        for (int j = 0; j < 8; ++j) {
            const int rl = ld_r + j * 4;
            const float4 v = *(const float4*)&smem[O_BASE + (w * 32 + rl) * LDS_STRIDE + ld_c];
            *(float4*)(out + (size_t)(row0 + rl) * T_LEN + k * TT + ld_c) = v;
        }
        // per-wave DS ops are in-order: next tile's orow writes cannot pass these reads.
    }
}

extern "C" void kernel_launch(void* const* d_in, const int* in_sizes, int n_in,
                              void* d_out, int out_size, void* d_ws, size_t ws_size,
                              hipStream_t stream)
{
    const float* strains = (const float*)d_in[0];
    const float* dts     = (const float*)d_in[1];
    float*       out     = (float*)d_out;

    const int B = in_sizes[0] / T_LEN;          // 16384
    const int grid = B / ROWS_PER_WG;           // 128 blocks x 4 independent waves

    maxwell_scan_kernel<<<grid, ROWS_PER_WG, 0, stream>>>(strains, dts, out);
}